// GraphConv_65197603553461
// MI455X (gfx1250) — compile-verified
//
#include <hip/hip_runtime.h>

#define N_NODES 100000
#define N_EDGES 600000
#define HIDDEN  128
#define LN_EPS  1e-5f

typedef __attribute__((ext_vector_type(2))) float v2f;
typedef __attribute__((ext_vector_type(8))) float v8f;

// ---------------- Kernel A: zero the aggregation buffer u ----------------
__global__ void gc_zero_kernel(float4* __restrict__ p, int n4) {
    int i = blockIdx.x * blockDim.x + threadIdx.x;
    int stride = gridDim.x * blockDim.x;
    float4 z = {0.f, 0.f, 0.f, 0.f};
    for (; i < n4; i += stride) p[i] = z;
}

// ---------------- Kernel B: edge gather + scatter-add --------------------
// One wave32 per edge; each lane moves 4 consecutive floats (float4 load,
// 4x global_atomic_add_f32 into the L2-resident u buffer).
__global__ __launch_bounds__(256) void gc_scatter_kernel(
    const float* __restrict__ h,
    const int*   __restrict__ src,
    const int*   __restrict__ dst,
    float*       __restrict__ u) {
    int edge = blockIdx.x * (blockDim.x >> 5) + (threadIdx.x >> 5);
    if (edge >= N_EDGES) return;
    int lane = threadIdx.x & 31;
    int s = src[edge];
    int d = dst[edge];
    const float4* hp = (const float4*)(h + (size_t)s * HIDDEN);
    float4 v = hp[lane];
    float* up = u + (size_t)d * HIDDEN + lane * 4;
    atomicAdd(up + 0, v.x);
    atomicAdd(up + 1, v.y);
    atomicAdd(up + 2, v.z);
    atomicAdd(up + 3, v.w);
}

// ------ Kernel C: z = ReLU(u @ W^T + b); out = LayerNorm(z + h) ----------
// Block = 256 threads = 8 waves, handles a 16-row tile.
// Wave w computes output columns [16w, 16w+16) via V_WMMA_F32_16X16X4_F32,
// looping K = 128 in steps of 4 (32 wmma per wave, 256 per block).
__global__ __launch_bounds__(256) void gc_gemm_ln_kernel(
    const float* __restrict__ u,
    const float* __restrict__ h,
    const float* __restrict__ W,
    const float* __restrict__ bias,
    const float* __restrict__ gamma,
    const float* __restrict__ beta,
    float*       __restrict__ out) {
    __shared__ float su[16][HIDDEN];       // u tile (A operand source)
    __shared__ float sh[16][HIDDEN];       // h tile (residual)
    __shared__ float sx[16][HIDDEN + 4];   // x = relu(z)+h (padded, 16B-aligned rows)
    __shared__ float srow[16][2];          // per-row mean, rstd

    const int tid  = threadIdx.x;
    const int row0 = blockIdx.x * 16;

    // Stage u and h tiles coalesced: 512 float4 each, 256 threads x 2.
    {
        const float4* gu = (const float4*)(u + (size_t)row0 * HIDDEN);
        const float4* gh = (const float4*)(h + (size_t)row0 * HIDDEN);
        float4* lu = (float4*)(&su[0][0]);
        float4* lh = (float4*)(&sh[0][0]);
        lu[tid]       = gu[tid];
        lu[tid + 256] = gu[tid + 256];
        lh[tid]       = gh[tid];
        lh[tid + 256] = gh[tid + 256];
    }
    __syncthreads();

    const int wave = tid >> 5;        // column-tile index 0..7
    const int lane = tid & 31;
    const int hlf  = lane >> 4;       // 0: K pair {0,1}; 1: K pair {2,3}
    const int lrow = lane & 15;       // A: M row / B: N column / D: N column
    const int n0   = wave * 16;
    const int col  = n0 + lrow;

    // C init = bias (D element i has column = col for all i)
    v8f acc;
    {
        float bv = bias[col];
        #pragma unroll
        for (int i = 0; i < 8; ++i) acc[i] = bv;
    }

    // z[m][n] = sum_k u[m][k] * W[n][k]  ->  A = u tile, B[k][n] = W[n][k]
    const float* wrow = W + (size_t)col * HIDDEN;
    #pragma unroll 4
    for (int ks = 0; ks < 32; ++ks) {
        int k = ks * 4 + 2 * hlf;
        v2f a, bw;
        a.x  = su[lrow][k];
        a.y  = su[lrow][k + 1];
        bw.x = wrow[k];
        bw.y = wrow[k + 1];
        acc = __builtin_amdgcn_wmma_f32_16x16x4_f32(
            /*neg_a=*/false, a, /*neg_b=*/false, bw,
            /*c_mod=*/(short)0, acc, /*reuse_a=*/false, /*reuse_b=*/false);
    }

    // ReLU + residual into LDS. D layout: VGPR i -> M = i + 8*hlf, N = lrow.
    #pragma unroll
    for (int i = 0; i < 8; ++i) {
        int r = i + 8 * hlf;
        float z = acc[i];
        z = z > 0.f ? z : 0.f;
        sx[r][col] = z + sh[r][col];
    }
    __syncthreads();

    // LayerNorm: 16 threads per row (contiguous 16-lane groups within a wave32),
    // each covers 8 elements; reduce with shfl_xor (masks 1,2,4,8 stay in-group).
    const int r   = tid >> 4;
    const int seg = tid & 15;
    {
        float s = 0.f, ss = 0.f;
        #pragma unroll
        for (int j = 0; j < 8; ++j) {
            float v = sx[r][seg * 8 + j];
            s  += v;
            ss += v * v;
        }
        #pragma unroll
        for (int m = 1; m < 16; m <<= 1) {
            s  += __shfl_xor(s,  m, 32);
            ss += __shfl_xor(ss, m, 32);
        }
        if (seg == 0) {
            float mean = s * (1.f / 128.f);
            float var  = ss * (1.f / 128.f) - mean * mean;
            srow[r][0] = mean;
            srow[r][1] = rsqrtf(var + LN_EPS);
        }
    }
    __syncthreads();

    // Normalize + affine, vectorized float4 stores.
    {
        float mean = srow[r][0];
        float rstd = srow[r][1];
        float4* o4 = (float4*)(out + (size_t)(row0 + r) * HIDDEN);
        const float4* g4 = (const float4*)gamma;
        const float4* b4 = (const float4*)beta;
        #pragma unroll
        for (int q = 0; q < 2; ++q) {
            int c4 = seg * 2 + q;
            float4 xv = *(const float4*)&sx[r][c4 * 4];
            float4 gv = g4[c4];
            float4 bv = b4[c4];
            float4 ov;
            ov.x = (xv.x - mean) * rstd * gv.x + bv.x;
            ov.y = (xv.y - mean) * rstd * gv.y + bv.y;
            ov.z = (xv.z - mean) * rstd * gv.z + bv.z;
            ov.w = (xv.w - mean) * rstd * gv.w + bv.w;
            o4[c4] = ov;
        }
    }
}

extern "C" void kernel_launch(void* const* d_in, const int* in_sizes, int n_in,
                              void* d_out, int out_size, void* d_ws, size_t ws_size,
                              hipStream_t stream) {
    const float* h     = (const float*)d_in[0];
    const int*   src   = (const int*)  d_in[1];
    const int*   dst   = (const int*)  d_in[2];
    const float* W     = (const float*)d_in[3];
    const float* bias  = (const float*)d_in[4];
    const float* gamma = (const float*)d_in[5];
    const float* beta  = (const float*)d_in[6];
    float* out = (float*)d_out;
    float* u   = (float*)d_ws;   // N_NODES * HIDDEN floats = 51.2 MB scratch

    const int n4 = (N_NODES * HIDDEN) / 4;
    gc_zero_kernel<<<2048, 256, 0, stream>>>((float4*)u, n4);
    gc_scatter_kernel<<<(N_EDGES + 7) / 8, 256, 0, stream>>>(h, src, dst, u);
    gc_gemm_ln_kernel<<<N_NODES / 16, 256, 0, stream>>>(u, h, W, bias, gamma, beta, out);
}